// Filter_1133871366182
// MI455X (gfx1250) — compile-verified
//
#include <hip/hip_runtime.h>

typedef __attribute__((ext_vector_type(2))) float v2f;
typedef __attribute__((ext_vector_type(8))) float v8f;

// Out[b,l,c,d] = u_c * a_d + v_c * b_d, realized as a K=4 (zero-padded K=2)
// f32 WMMA per 16x16 output tile. One wave per tile.
__global__ __launch_bounds__(256) void Filter_rank2_wmma_kernel(
    const float* __restrict__ x,     // (32,1024,2) f32
    const float* __restrict__ filt,  // (1,2,64,64) f32
    float* __restrict__ out)         // (32,2,1024,1024) f32
{
  const int wave = (blockIdx.x << 3) | (threadIdx.x >> 5);
  const int lane = threadIdx.x & 31;
  const int r    = lane & 15;
  const bool lo  = lane < 16;

  const int dt   = wave & 63;          // d tile (cols)
  const int ct   = (wave >> 6) & 63;   // c tile (rows)
  const int pair = wave >> 12;         // b*2 + l, 0..63
  const int b    = pair >> 1;
  const int l    = pair & 1;

  // 2x2 effective filter block at s0 = 32 - l (uniform -> scalar loads)
  const int s0 = 32 - l;
  const float* F = filt + l * 4096;
  const float F00 = F[s0 * 64 + s0];
  const float F01 = F[s0 * 64 + s0 + 1];
  const float F10 = F[(s0 + 1) * 64 + s0];
  const float F11 = F[(s0 + 1) * 64 + s0 + 1];

  // Per-row (a,b) pairs: x[(b*1024 + row)*2 + {0,1}] as 8-byte loads
  const v2f* __restrict__ x2 = (const v2f*)x;
  const int xbase = b << 10;
  const v2f xc = x2[xbase + (ct << 4) + r];  // rows of this c-tile
  const v2f xd = x2[xbase + (dt << 4) + r];  // rows of this d-tile

  // A fragment (16x4 f32, 2 VGPRs): lanes 0-15 -> K=0 (u), K=1 (v);
  // lanes 16-31 -> K=2, K=3 (zero padding of the rank-2 update)
  v2f A;
  A.x = lo ? (xc.x * F00 + xc.y * F10) : 0.0f;  // u_c
  A.y = lo ? (xc.x * F01 + xc.y * F11) : 0.0f;  // v_c

  // B fragment (4x16 f32, 2 VGPRs): VGPR0 = row K=0 (a_d) / K=2 (0),
  // VGPR1 = row K=1 (b_d) / K=3 (0)
  v2f B;
  B.x = lo ? xd.x : 0.0f;
  B.y = lo ? xd.y : 0.0f;

  v8f C = {};
  // (neg_a, A, neg_b, B, c_mod, C, reuse_a, reuse_b)
  v8f D = __builtin_amdgcn_wmma_f32_16x16x4_f32(
      false, A, false, B, (short)0, C, false, false);

  // C/D layout: VGPR i -> row M = i + (lane>=16 ? 8 : 0), col N = lane%16.
  // Output is 256MB streaming (> 192MB L2): use non-temporal stores.
  float* o = out + ((size_t)pair << 20)
                 + ((size_t)((ct << 4) + (lo ? 0 : 8)) << 10)
                 + (size_t)((dt << 4) + r);
#pragma unroll
  for (int i = 0; i < 8; ++i)
    __builtin_nontemporal_store(D[i], o + ((size_t)i << 10));
}

extern "C" void kernel_launch(void* const* d_in, const int* in_sizes, int n_in,
                              void* d_out, int out_size, void* d_ws, size_t ws_size,
                              hipStream_t stream) {
  const float* x    = (const float*)d_in[0];   // 32*1024*2
  const float* filt = (const float*)d_in[1];   // 1*2*64*64
  float* out = (float*)d_out;                  // 32*2*1024*1024

  // 64 (b,l) pairs * 64 * 64 tiles = 262144 waves; 8 waves per 256-thread block
  dim3 grid(32768), block(256);
  hipLaunchKernelGGL(Filter_rank2_wmma_kernel, grid, block, 0, stream,
                     x, filt, out);
}